// TokenPruner_38860864094847
// MI455X (gfx1250) — compile-verified
//
#include <hip/hip_runtime.h>
#include <hip/hip_bf16.h>

// Problem constants (from reference): B=2, H=16, S=2048, D=1024
#define BB 2
#define HH 16
#define SS 2048
#define DD 1024
#define NEGV 10000.0f

// WMMA colsum tiling: each wave owns COLS columns x ROWCHUNK rows of one (b,h) plane
#define COLS 64
#define ROWCHUNK 512

typedef float v2f __attribute__((ext_vector_type(2)));
typedef float v4f __attribute__((ext_vector_type(4)));
typedef float v8f __attribute__((ext_vector_type(8)));

// ---------------------------------------------------------------------------
// Stage 0: zero the score accumulator (must happen every launch: atomics below)
// ---------------------------------------------------------------------------
__global__ void tp_zero(float* __restrict__ score, int n) {
    int i = blockIdx.x * blockDim.x + threadIdx.x;
    if (i < n) score[i] = 0.0f;
}

// ---------------------------------------------------------------------------
// Stage 1: score[b][j] += hm[h] * sum_i probs[b,h,i,j], via WMMA with A = ones.
// D = A(16x4 ones) x B(4x16 tile of P) + C  =>  every row of C = column sums.
// 8 coalesced b32 loads (1 KB) feed 4 WMMAs per 4-row step; memory-bound.
// probs is a one-pass 512 MiB stream -> non-temporal loads (don't pollute L2).
// EXEC is all-ones for the whole WMMA loop (no divergence until the epilogue).
// ---------------------------------------------------------------------------
__global__ void tp_colsum_wmma(const float* __restrict__ probs,
                               const float* __restrict__ hm,
                               float* __restrict__ score) {
    const int lane = threadIdx.x & 31;
    const int wave = blockIdx.x * (blockDim.x >> 5) + (threadIdx.x >> 5);

    // wave id bits: rc(2) | ct(5) | h(4) | b(1)  -> 4096 waves total
    const int rc  = wave & 3;          // row chunk: S/ROWCHUNK = 4
    int tmp       = wave >> 2;
    const int ct  = tmp & 31;          // col tile:  S/COLS = 32
    tmp >>= 5;
    const int h   = tmp & 15;
    const int b   = tmp >> 4;

    const int col0 = ct * COLS;
    const int row0 = rc * ROWCHUNK;
    const int half = lane >> 4;        // which pair of K rows this lane supplies
    const int colL = lane & 15;        // lane <-> column mapping (B/C/D stripe N over lanes)

    const float* base = probs
        + (((size_t)(b * HH + h) * SS + (size_t)(row0 + 2 * half)) * SS)
        + col0 + colL;

    v2f a; a.x = 1.0f; a.y = 1.0f;     // ones matrix
    v8f acc0 = {}, acc1 = {}, acc2 = {}, acc3 = {};

    for (int k = 0; k < ROWCHUNK; k += 4) {
        const float* p = base + (size_t)k * SS;
        v2f b0; b0.x = __builtin_nontemporal_load(p +  0);
                b0.y = __builtin_nontemporal_load(p + SS);
        v2f b1; b1.x = __builtin_nontemporal_load(p + 16);
                b1.y = __builtin_nontemporal_load(p + SS + 16);
        v2f b2; b2.x = __builtin_nontemporal_load(p + 32);
                b2.y = __builtin_nontemporal_load(p + SS + 32);
        v2f b3; b3.x = __builtin_nontemporal_load(p + 48);
                b3.y = __builtin_nontemporal_load(p + SS + 48);
        acc0 = __builtin_amdgcn_wmma_f32_16x16x4_f32(false, a, false, b0, (short)0, acc0, false, false);
        acc1 = __builtin_amdgcn_wmma_f32_16x16x4_f32(false, a, false, b1, (short)0, acc1, false, false);
        acc2 = __builtin_amdgcn_wmma_f32_16x16x4_f32(false, a, false, b2, (short)0, acc2, false, false);
        acc3 = __builtin_amdgcn_wmma_f32_16x16x4_f32(false, a, false, b3, (short)0, acc3, false, false);
    }

    // Row M=0 of C (VGPR0, lanes 0..15) holds the column sums.
    const float w = hm[h];
    if (lane < 16) {
        float* s = score + (size_t)b * SS + col0 + colL;
        atomicAdd(s +  0, acc0[0] * w);
        atomicAdd(s + 16, acc1[0] * w);
        atomicAdd(s + 32, acc2[0] * w);
        atomicAdd(s + 48, acc3[0] * w);
    }
}

// ---------------------------------------------------------------------------
// Stage 2: per batch (one wave32): finalize score, keep-test, stable stream
// compaction via ballot+popcount, write padded_idx and the attention-mask out.
// ---------------------------------------------------------------------------
__global__ void tp_compact(const float* __restrict__ score,
                           const float* __restrict__ hm,
                           const float* __restrict__ amask,
                           const float* __restrict__ thr,
                           int* __restrict__ idx,
                           float* __restrict__ am_out) {
    const int b = blockIdx.x;
    const int lane = threadIdx.x;      // blockDim = 32, one full wave

    float hmsum = 0.0f;
    #pragma unroll
    for (int h = 0; h < HH; ++h) hmsum = hmsum + hm[h];
    const float t = thr[0];

    // zero-fill padded_idx (pad slots stay 0); stores from the same wave are
    // kept in order with later scatter stores.
    for (int j = lane; j < SS; j += 32) idx[b * SS + j] = 0;

    unsigned base = 0;
    for (int j0 = 0; j0 < SS; j0 += 32) {
        const int j = j0 + lane;
        float sc = score[b * SS + j] / hmsum + (j == 0 ? 100.0f : 0.0f);
        const bool keep = sc > t;
        const unsigned m = (unsigned)__ballot(keep);   // wave32: low 32 bits
        if (keep) {
            const int pos = (int)(base + __popc(m & ((1u << lane) - 1u)));
            idx[b * SS + pos] = j;
        }
        base += __popc(m);
    }

    // attention-mask output: gather + NEG fixups (store->load same wave is ordered)
    for (int p = lane; p < SS; p += 32) {
        const int id = idx[b * SS + p];
        float v = amask[b * SS + id];
        if (id == 0) v -= NEGV;        // pad slots (and the real CLS)
        if (p == 0)  v += NEGV;        // slot 0 is real [CLS]
        am_out[b * SS + p] = v;
    }
}

// ---------------------------------------------------------------------------
// Stage 3: gather hidden_states rows by padded_idx, 128-bit coalesced.
// grid = B*S blocks, 256 threads; D/4 = 256 v4f per row.
// Loads stay RT (pad slots re-read row 0 -> cache-friendly); stores are NT
// (write-once 16 MiB stream). v4f is a clang ext-vector so the NT builtin
// accepts it (HIP's float4 struct does not).
// ---------------------------------------------------------------------------
__global__ void tp_gather(const v4f* __restrict__ hsrc,
                          const int* __restrict__ idx,
                          v4f* __restrict__ out) {
    const int bs = blockIdx.x;         // b*S + p
    const int b  = bs >> 11;           // /2048
    const int id = idx[bs];
    const size_t src = ((size_t)b * SS + (size_t)id) * (DD / 4) + threadIdx.x;
    const size_t dst = (size_t)bs * (DD / 4) + threadIdx.x;
    const v4f v = hsrc[src];
    __builtin_nontemporal_store(v, out + dst);
}

extern "C" void kernel_launch(void* const* d_in, const int* in_sizes, int n_in,
                              void* d_out, int out_size, void* d_ws, size_t ws_size,
                              hipStream_t stream) {
    const float* hidden = (const float*)d_in[0];   // [B,S,D]
    const float* probs  = (const float*)d_in[1];   // [B,H,S,S]
    const float* hm     = (const float*)d_in[2];   // [H]
    const float* amask  = (const float*)d_in[3];   // [B,1,1,S] -> [B,S]
    const float* thr    = (const float*)d_in[4];   // [1]

    float* out    = (float*)d_out;
    float* hs_out = out;                               // B*S*D floats
    float* am_out = out + (size_t)BB * SS * DD;        // B*S floats

    float* score = (float*)d_ws;                       // B*S floats
    int*   idx   = (int*)((char*)d_ws + (size_t)BB * SS * sizeof(float)); // B*S ints

    // Stage 0: zero score accumulator
    tp_zero<<<(BB * SS + 255) / 256, 256, 0, stream>>>(score, BB * SS);

    // Stage 1: WMMA column-sum reduction over the 512 MiB probs tensor
    const int waves  = BB * HH * (SS / COLS) * (SS / ROWCHUNK); // 4096
    const int blocks = waves / 8;                               // 256 thr = 8 waves
    tp_colsum_wmma<<<blocks, 256, 0, stream>>>(probs, hm, score);

    // Stage 2: keep-test + stable compaction + attention-mask output
    tp_compact<<<BB, 32, 0, stream>>>(score, hm, amask, thr, idx, am_out);

    // Stage 3: hidden_states gather
    tp_gather<<<BB * SS, 256, 0, stream>>>((const v4f*)hidden, idx, (v4f*)hs_out);
}